// GraphDecoder_35699768164478
// MI455X (gfx1250) — compile-verified
//
#include <hip/hip_runtime.h>
#include <hip/hip_bf16.h>
#include <math.h>

#define B_    64
#define H_    512
#define EMB_  512
#define V_    32000
#define SENC_ 512
#define SSEQ_ 512
#define ENC2_ 512
#define OOV_  50
#define VEXT_ (V_ + OOV_)          // 32050
#define NEGINF_ 1000000000000.0f
#define LDSPITCH_ 516              // 512 + 4 floats pad -> conflict-free fragment reads

typedef __attribute__((ext_vector_type(2))) float v2f;
typedef __attribute__((ext_vector_type(8))) float v8f;

// ---------- helpers ----------
__device__ __forceinline__ float wredsum(float v) {
#pragma unroll
  for (int m = 16; m >= 1; m >>= 1) v += __shfl_xor(v, m, 32);
  return v;
}
__device__ __forceinline__ float wredmax(float v) {
#pragma unroll
  for (int m = 16; m >= 1; m >>= 1) v = fmaxf(v, __shfl_xor(v, m, 32));
  return v;
}
__device__ __forceinline__ unsigned f2ord(float f) {
  unsigned u = __float_as_uint(f);
  return (u & 0x80000000u) ? ~u : (u | 0x80000000u);
}
__device__ __forceinline__ float ord2f(unsigned u) {
  return (u & 0x80000000u) ? __uint_as_float(u & 0x7fffffffu)
                           : __uint_as_float(~u);
}
__device__ __forceinline__ float sigmoidf(float x) { return 1.0f / (1.0f + expf(-x)); }

// ---------- K0: init copy buffer (ordered -INF), OOV zeros in d_out, seqE=0 ----------
__global__ void k_init(unsigned* copybuf, float* out, float* seqE) {
  int idx = blockIdx.x * blockDim.x + threadIdx.x;
  if (idx < B_ * VEXT_) {
    copybuf[idx] = f2ord(-NEGINF_);
    int v = idx % VEXT_;
    if (v >= V_) out[idx] = 0.0f;   // extended-vocab zeros
  }
  if (idx < B_ * H_) seqE[idx] = 0.0f;
}

// ---------- K1: xcat = [embed[ids], prev_context]  (B x 1024) ----------
__global__ void k_xcat(const int* __restrict__ ids, const float* __restrict__ embed,
                       const float* __restrict__ prevctx, float* __restrict__ xcat) {
  int idx = blockIdx.x * blockDim.x + threadIdx.x;
  if (idx >= B_ * (EMB_ + H_)) return;
  int b = idx >> 10, i = idx & 1023;
  xcat[idx] = (i < EMB_) ? embed[(size_t)ids[b] * EMB_ + i]
                         : prevctx[(size_t)b * H_ + (i - EMB_)];
}

// ---------- K2: x = xcat @ W_reduce^T + b_reduce   (wave per (b,j)) ----------
__global__ void k_reduce(const float* __restrict__ xcat, const float* __restrict__ Wr,
                         const float* __restrict__ br, float* __restrict__ x) {
  int lane = threadIdx.x & 31, wv = threadIdx.x >> 5;
  int j = blockIdx.x * 8 + wv, b = blockIdx.y;
  const float* xr = xcat + (size_t)b * 1024;
  const float* wr = Wr + (size_t)j * 1024;
  float acc = 0.0f;
  for (int k = lane; k < 1024; k += 32) acc += xr[k] * wr[k];
  acc = wredsum(acc);
  if (lane == 0) x[(size_t)b * EMB_ + j] = acc + br[j];
}

// ---------- K3: gates = x@W_ih^T + h0@W_hh^T + b_ih + b_hh  (wave per (b,j)) ----------
__global__ void k_gates(const float* __restrict__ x, const float* __restrict__ h0,
                        const float* __restrict__ Wih, const float* __restrict__ Whh,
                        const float* __restrict__ bih, const float* __restrict__ bhh,
                        float* __restrict__ gates) {
  int lane = threadIdx.x & 31, wv = threadIdx.x >> 5;
  int j = blockIdx.x * 8 + wv, b = blockIdx.y;
  const float* xr = x + (size_t)b * EMB_;
  const float* hr = h0 + (size_t)b * H_;
  const float* wi = Wih + (size_t)j * EMB_;
  const float* wh = Whh + (size_t)j * H_;
  float acc = 0.0f;
  for (int k = lane; k < 512; k += 32) acc += xr[k] * wi[k] + hr[k] * wh[k];
  acc = wredsum(acc);
  if (lane == 0) gates[(size_t)b * 2048 + j] = acc + bih[j] + bhh[j];
}

// ---------- K4: LSTM cell elementwise ----------
__global__ void k_lstm(const float* __restrict__ gates, const float* __restrict__ c0,
                       float* __restrict__ h1, float* __restrict__ c1,
                       float* __restrict__ oh1, float* __restrict__ oc1) {
  int idx = blockIdx.x * blockDim.x + threadIdx.x;
  if (idx >= B_ * H_) return;
  int b = idx >> 9, h = idx & 511;
  const float* g = gates + (size_t)b * 2048;
  float i_ = sigmoidf(g[h]);
  float f_ = sigmoidf(g[h + 512]);
  float gg = tanhf(g[h + 1024]);
  float o_ = sigmoidf(g[h + 1536]);
  float c = f_ * c0[idx] + i_ * gg;
  float hh = o_ * tanhf(c);
  c1[idx] = c; h1[idx] = hh; oc1[idx] = c; oh1[idx] = hh;
}

// ---------- K5: q[b,j] = sum_h h1[b,h] * W_key[h,j]   (factorized attention) ----------
__global__ void k_q(const float* __restrict__ h1, const float* __restrict__ Wkey,
                    float* __restrict__ q) {
  int idx = blockIdx.x * blockDim.x + threadIdx.x;
  if (idx >= B_ * H_) return;
  int b = idx >> 9, j = idx & 511;
  const float* hr = h1 + (size_t)b * H_;
  float acc = 0.0f;
  for (int h = 0; h < H_; ++h) acc += hr[h] * Wkey[(size_t)h * H_ + j];
  q[idx] = acc;
}

// ---------- K6: masked attention: energies, softmax, context (block per b) ----------
__global__ void k_attn(const float* __restrict__ enc, const float* __restrict__ q,
                       const float* __restrict__ bkey, const float* __restrict__ h1,
                       float* __restrict__ ctx, float* __restrict__ octx,
                       float* __restrict__ oattn) {
  __shared__ float sQ[SENC_];
  __shared__ float sA[SENC_];
  __shared__ float sRed[8];
  __shared__ float sScal;
  int b = blockIdx.x, t = threadIdx.x;
  int lane = t & 31, wv = t >> 5;
  for (int i = t; i < H_; i += 256) sQ[i] = q[(size_t)b * H_ + i];
  // e0 = b_key . h1[b]
  float p = 0.0f;
  for (int i = t; i < H_; i += 256) p += bkey[i] * h1[(size_t)b * H_ + i];
  p = wredsum(p);
  if (lane == 0) sRed[wv] = p;
  __syncthreads();
  if (t == 0) { float s = 0; for (int i = 0; i < 8; ++i) s += sRed[i]; sScal = s; }
  __syncthreads();
  float e0 = sScal;
  // energies (wave per row)
  for (int s = wv; s < SENC_; s += 8) {
    const float* er = enc + ((size_t)b * SENC_ + s) * H_;
    float d = 0.0f, rs = 0.0f;
    for (int k = lane; k < H_; k += 32) { float e = er[k]; d += e * sQ[k]; rs += e; }
    d = wredsum(d); rs = wredsum(rs);
    if (lane == 0) sA[s] = (rs == 0.0f) ? 1e-12f : (d + e0);
  }
  __syncthreads();
  // softmax max
  float mx = -INFINITY;
  for (int i = t; i < SENC_; i += 256) mx = fmaxf(mx, sA[i]);
  mx = wredmax(mx);
  if (lane == 0) sRed[wv] = mx;
  __syncthreads();
  if (t == 0) { float m = sRed[0]; for (int i = 1; i < 8; ++i) m = fmaxf(m, sRed[i]); sScal = m; }
  __syncthreads();
  mx = sScal;
  float sm = 0.0f;
  for (int i = t; i < SENC_; i += 256) { float e = expf(sA[i] - mx); sA[i] = e; sm += e; }
  sm = wredsum(sm);
  if (lane == 0) sRed[wv] = sm;
  __syncthreads();
  if (t == 0) { float s = 0; for (int i = 0; i < 8; ++i) s += sRed[i]; sScal = s; }
  __syncthreads();
  float inv = 1.0f / sScal;
  for (int i = t; i < SENC_; i += 256) {
    float a = sA[i] * inv; sA[i] = a; oattn[(size_t)b * SENC_ + i] = a;
  }
  __syncthreads();
  // context
  for (int h = t; h < H_; h += 256) {
    const float* eb = enc + (size_t)b * SENC_ * H_ + h;
    float acc = 0.0f;
    for (int s = 0; s < SENC_; ++s) acc += sA[s] * eb[(size_t)s * H_];
    ctx[(size_t)b * H_ + h] = acc;
    octx[(size_t)b * H_ + h] = acc;
  }
}

// ---------- K7: combined = tanh([h1,ctx] @ W_comb^T + b_comb)  (wave per (b,j)) ----------
__global__ void k_comb(const float* __restrict__ h1, const float* __restrict__ ctx,
                       const float* __restrict__ Wc, const float* __restrict__ bc,
                       float* __restrict__ comb) {
  int lane = threadIdx.x & 31, wv = threadIdx.x >> 5;
  int j = blockIdx.x * 8 + wv, b = blockIdx.y;
  const float* wr = Wc + (size_t)j * 1024;
  float acc = 0.0f;
  for (int k = lane; k < 1024; k += 32) {
    float xv = (k < 512) ? h1[(size_t)b * H_ + k] : ctx[(size_t)b * H_ + (k - 512)];
    acc += xv * wr[k];
  }
  acc = wredsum(acc);
  if (lane == 0) comb[(size_t)b * H_ + j] = tanhf(acc + bc[j]);
}

// ---------- K8: logits GEMM via f32 WMMA ----------
// wave per 16-column tile of W_out; covers all M=64 with 4 accumulators so
// each B fragment (W_out, HBM-streamed) is loaded exactly once chip-wide.
__global__ void k_logits(const float* __restrict__ comb, const float* __restrict__ Wout,
                         const float* __restrict__ bout, float* __restrict__ out) {
  int lane = threadIdx.x & 31, wv = threadIdx.x >> 5;
  int w = blockIdx.x * 8 + wv;          // 0..1999 n-tiles
  if (w >= V_ / 16) return;
  int v0 = w * 16;
  int r16 = lane & 15, half = lane >> 4;
  const float* A0 = comb + (size_t)(r16) * H_;        // m-tile 0 rows
  const float* A1 = comb + (size_t)(16 + r16) * H_;
  const float* A2 = comb + (size_t)(32 + r16) * H_;
  const float* A3 = comb + (size_t)(48 + r16) * H_;
  const float* Bb = Wout + (size_t)(v0 + r16) * H_;   // B: W_out^T tile, n = lane&15
  v8f c0 = {}, c1 = {}, c2 = {}, c3 = {};
  for (int k = 0; k < H_; k += 4) {
    int ko = k + 2 * half;
    v2f bb = *(const v2f*)(Bb + ko);
    v2f a0 = *(const v2f*)(A0 + ko);
    v2f a1 = *(const v2f*)(A1 + ko);
    v2f a2 = *(const v2f*)(A2 + ko);
    v2f a3 = *(const v2f*)(A3 + ko);
    c0 = __builtin_amdgcn_wmma_f32_16x16x4_f32(false, a0, false, bb, (short)0, c0, false, false);
    c1 = __builtin_amdgcn_wmma_f32_16x16x4_f32(false, a1, false, bb, (short)0, c1, false, false);
    c2 = __builtin_amdgcn_wmma_f32_16x16x4_f32(false, a2, false, bb, (short)0, c2, false, false);
    c3 = __builtin_amdgcn_wmma_f32_16x16x4_f32(false, a3, false, bb, (short)0, c3, false, false);
  }
  float bo = bout[v0 + r16];
#pragma unroll
  for (int r = 0; r < 8; ++r) {
    int m = r + 8 * half;
    out[(size_t)(m) * VEXT_ + v0 + r16]      = c0[r] + bo;
    out[(size_t)(m + 16) * VEXT_ + v0 + r16] = c1[r] + bo;
    out[(size_t)(m + 32) * VEXT_ + v0 + r16] = c2[r] + bo;
    out[(size_t)(m + 48) * VEXT_ + v0 + r16] = c3[r] + bo;
  }
}

// ---------- K9: seq energies via f32 WMMA + tanh.dot epilogue ----------
// Block (8 waves) owns (b, 32-row s-block). A panel (32x512) staged ONCE into
// LDS via gfx1250 async copy (ASYNCcnt), then each wave (one 64-col h-group)
// does 2 s-tiles x 4 h-tiles register blocking. Partials -> [B,S,8].
__global__ void k_seqE(const float* __restrict__ seqs, const float* __restrict__ Wsk,
                       const float* __restrict__ h1, float* __restrict__ partE) {
  __shared__ float sA[32 * LDSPITCH_];
  int t = threadIdx.x;
  int lane = t & 31, wv = t >> 5;
  int blk = blockIdx.x;                 // 0..1023
  int b = blk >> 4;                     // / 16
  int sb = blk & 15;                    // s-block
  int s0 = sb * 32;
  int hg = wv;                          // h-group 0..7 -> 64 columns
  int h0 = hg * 64;
  int r16 = lane & 15, half = lane >> 4;

  // ---- stage A panel: 32 rows x 512 f32, 16B per lane per async op ----
  {
    const float* gbase = seqs + ((size_t)b * SSEQ_ + s0) * ENC2_;
#pragma unroll
    for (int i = 0; i < 16; ++i) {
      int c = t + 256 * i;              // 4096 float4 chunks
      int row = c >> 5;                 // 32 float4 per row
      int col = (c & 31) * 4;
      unsigned lds_off = (unsigned)(size_t)(&sA[row * LDSPITCH_ + col]);
      const float* g = gbase + (size_t)row * ENC2_ + col;
      asm volatile("global_load_async_to_lds_b128 %0, %1, off"
                   :: "v"(lds_off), "v"(g) : "memory");
    }
    asm volatile("s_wait_asynccnt 0x0" ::: "memory");
    __syncthreads();
  }

  const float* B0 = Wsk + (size_t)(h0 + r16) * ENC2_;
  const float* aLo = &sA[(size_t)r16 * LDSPITCH_];
  const float* aHi = &sA[(size_t)(16 + r16) * LDSPITCH_];
  v8f c00 = {}, c01 = {}, c02 = {}, c03 = {};   // s-tile 0, h-tiles 0..3
  v8f c10 = {}, c11 = {}, c12 = {}, c13 = {};   // s-tile 1, h-tiles 0..3
  for (int k = 0; k < ENC2_; k += 4) {
    int ko = k + 2 * half;
    v2f alo = *(const v2f*)(aLo + ko);
    v2f ahi = *(const v2f*)(aHi + ko);
    v2f b0 = *(const v2f*)(B0 + ko);
    v2f b1 = *(const v2f*)(B0 + 16 * ENC2_ + ko);
    v2f b2 = *(const v2f*)(B0 + 32 * ENC2_ + ko);
    v2f b3 = *(const v2f*)(B0 + 48 * ENC2_ + ko);
    c00 = __builtin_amdgcn_wmma_f32_16x16x4_f32(false, alo, false, b0, (short)0, c00, false, false);
    c01 = __builtin_amdgcn_wmma_f32_16x16x4_f32(false, alo, false, b1, (short)0, c01, false, false);
    c02 = __builtin_amdgcn_wmma_f32_16x16x4_f32(false, alo, false, b2, (short)0, c02, false, false);
    c03 = __builtin_amdgcn_wmma_f32_16x16x4_f32(false, alo, false, b3, (short)0, c03, false, false);
    c10 = __builtin_amdgcn_wmma_f32_16x16x4_f32(false, ahi, false, b0, (short)0, c10, false, false);
    c11 = __builtin_amdgcn_wmma_f32_16x16x4_f32(false, ahi, false, b1, (short)0, c11, false, false);
    c12 = __builtin_amdgcn_wmma_f32_16x16x4_f32(false, ahi, false, b2, (short)0, c12, false, false);
    c13 = __builtin_amdgcn_wmma_f32_16x16x4_f32(false, ahi, false, b3, (short)0, c13, false, false);
  }
  const float* hr = h1 + (size_t)b * H_ + h0;
  float o0 = hr[r16], o1 = hr[16 + r16], o2 = hr[32 + r16], o3 = hr[48 + r16];
#pragma unroll
  for (int r = 0; r < 8; ++r) {
    float vlo = tanhf(c00[r]) * o0 + tanhf(c01[r]) * o1 +
                tanhf(c02[r]) * o2 + tanhf(c03[r]) * o3;
    float vhi = tanhf(c10[r]) * o0 + tanhf(c11[r]) * o1 +
                tanhf(c12[r]) * o2 + tanhf(c13[r]) * o3;
    vlo += __shfl_xor(vlo, 1, 32); vlo += __shfl_xor(vlo, 2, 32);
    vlo += __shfl_xor(vlo, 4, 32); vlo += __shfl_xor(vlo, 8, 32);
    vhi += __shfl_xor(vhi, 1, 32); vhi += __shfl_xor(vhi, 2, 32);
    vhi += __shfl_xor(vhi, 4, 32); vhi += __shfl_xor(vhi, 8, 32);
    if (r16 == 0) {
      int srow = s0 + r + 8 * half;
      partE[((size_t)b * SSEQ_ + srow) * 8 + hg] = vlo;
      partE[((size_t)b * SSEQ_ + srow + 16) * 8 + hg] = vhi;
    }
  }
}

// ---------- K10: deterministic reduce of 8 h-group partials ----------
__global__ void k_seqred(const float* __restrict__ partE, float* __restrict__ seqE) {
  int idx = blockIdx.x * blockDim.x + threadIdx.x;
  if (idx >= B_ * SSEQ_) return;
  const float* p = partE + (size_t)idx * 8;
  float s = 0.0f;
#pragma unroll
  for (int g = 0; g < 8; ++g) s += p[g];
  seqE[idx] = s;
}

// ---------- K11: masked scatter-max into extended-vocab copy buffer ----------
__global__ void k_scatter(const float* __restrict__ seqE,
                          const unsigned char* __restrict__ mask,
                          const int* __restrict__ ext, unsigned* __restrict__ copybuf) {
  int b = blockIdx.x, s = threadIdx.x;
  float val = mask[(size_t)b * SSEQ_ + s] ? 1e-12f : seqE[(size_t)b * SSEQ_ + s];
  int idx = ext[(size_t)b * SSEQ_ + s];
  atomicMax(copybuf + (size_t)b * VEXT_ + idx, f2ord(val));
}

// ---------- K12: combine copy + extended, then log_softmax over VEXT (block per b) ----------
__global__ void k_final(float* __restrict__ out, const unsigned* __restrict__ copybuf) {
  __shared__ float sRed[8];
  __shared__ float sScal;
  int b = blockIdx.x, t = threadIdx.x;
  int lane = t & 31, wv = t >> 5;
  float* row = out + (size_t)b * VEXT_;
  const unsigned* crow = copybuf + (size_t)b * VEXT_;
  float mx = -INFINITY;
  for (int v = t; v < VEXT_; v += 256) {
    float ext = row[v];
    float cp = ord2f(crow[v]);
    float cv = (cp == -NEGINF_) ? 0.0f : cp;
    float o = ext + cv;
    if (o == 0.0f) o = -NEGINF_;
    row[v] = o;
    mx = fmaxf(mx, o);
  }
  mx = wredmax(mx);
  if (lane == 0) sRed[wv] = mx;
  __syncthreads();
  if (t == 0) { float m = sRed[0]; for (int i = 1; i < 8; ++i) m = fmaxf(m, sRed[i]); sScal = m; }
  __syncthreads();
  mx = sScal;
  float sm = 0.0f;
  for (int v = t; v < VEXT_; v += 256) sm += expf(row[v] - mx);
  sm = wredsum(sm);
  if (lane == 0) sRed[wv] = sm;
  __syncthreads();
  if (t == 0) { float s = 0; for (int i = 0; i < 8; ++i) s += sRed[i]; sScal = s; }
  __syncthreads();
  float lg = mx + logf(sScal);
  for (int v = t; v < VEXT_; v += 256) row[v] = row[v] - lg;
}

// ---------------- host launcher ----------------
extern "C" void kernel_launch(void* const* d_in, const int* in_sizes, int n_in,
                              void* d_out, int out_size, void* d_ws, size_t ws_size,
                              hipStream_t stream) {
  const int*   ids    = (const int*)d_in[0];
  const float* prevc  = (const float*)d_in[1];
  const float* h0     = (const float*)d_in[2];
  const float* c0     = (const float*)d_in[3];
  const float* enc    = (const float*)d_in[4];
  const float* seqs   = (const float*)d_in[5];
  const unsigned char* mask = (const unsigned char*)d_in[6];
  const int*   ext    = (const int*)d_in[7];
  const float* embed  = (const float*)d_in[8];
  const float* Wr     = (const float*)d_in[9];
  const float* br     = (const float*)d_in[10];
  const float* Wkey   = (const float*)d_in[11];
  const float* bkey   = (const float*)d_in[12];
  const float* Wc     = (const float*)d_in[13];
  const float* bc     = (const float*)d_in[14];
  const float* Wsk    = (const float*)d_in[15];
  const float* Wih    = (const float*)d_in[16];
  const float* Whh    = (const float*)d_in[17];
  const float* bih    = (const float*)d_in[18];
  const float* bhh    = (const float*)d_in[19];
  const float* Wout   = (const float*)d_in[20];
  const float* bout   = (const float*)d_in[21];

  float* out   = (float*)d_out;                    // [B, VEXT]
  float* octx  = out + (size_t)B_ * VEXT_;         // [B, H]
  float* oh1   = octx + (size_t)B_ * H_;           // [1, B, H]
  float* oc1   = oh1 + (size_t)B_ * H_;            // [1, B, H]
  float* oattn = oc1 + (size_t)B_ * H_;            // [B, S_ENC]

  // workspace carve-up (floats)
  float* ws    = (float*)d_ws;
  float* xcat  = ws;                                // 64*1024
  float* x     = xcat + (size_t)B_ * 1024;          // 64*512
  float* gates = x + (size_t)B_ * EMB_;             // 64*2048
  float* h1    = gates + (size_t)B_ * 2048;         // 64*512
  float* c1    = h1 + (size_t)B_ * H_;              // 64*512
  float* q     = c1 + (size_t)B_ * H_;              // 64*512
  float* ctx   = q + (size_t)B_ * H_;               // 64*512
  float* comb  = ctx + (size_t)B_ * H_;             // 64*512
  float* seqE  = comb + (size_t)B_ * H_;            // 64*512
  float* partE = seqE + (size_t)B_ * SSEQ_;         // 64*512*8
  unsigned* copybuf = (unsigned*)(partE + (size_t)B_ * SSEQ_ * 8); // 64*32050

  k_init<<<(B_ * VEXT_ + 255) / 256, 256, 0, stream>>>(copybuf, out, seqE);
  k_xcat<<<(B_ * 1024 + 255) / 256, 256, 0, stream>>>(ids, embed, prevc, xcat);
  k_reduce<<<dim3(EMB_ / 8, B_), 256, 0, stream>>>(xcat, Wr, br, x);
  k_gates<<<dim3(2048 / 8, B_), 256, 0, stream>>>(x, h0, Wih, Whh, bih, bhh, gates);
  k_lstm<<<(B_ * H_ + 255) / 256, 256, 0, stream>>>(gates, c0, h1, c1, oh1, oc1);
  k_q<<<(B_ * H_ + 255) / 256, 256, 0, stream>>>(h1, Wkey, q);
  k_attn<<<B_, 256, 0, stream>>>(enc, q, bkey, h1, ctx, octx, oattn);
  k_comb<<<dim3(H_ / 8, B_), 256, 0, stream>>>(h1, ctx, Wc, bc, comb);
  k_logits<<<(V_ / 16 + 7) / 8, 256, 0, stream>>>(comb, Wout, bout, out);
  k_seqE<<<B_ * (SSEQ_ / 32), 256, 0, stream>>>(seqs, Wsk, h1, partE);
  k_seqred<<<(B_ * SSEQ_ + 255) / 256, 256, 0, stream>>>(partE, seqE);
  k_scatter<<<B_, SSEQ_, 0, stream>>>(seqE, mask, ext, copybuf);
  k_final<<<B_, 256, 0, stream>>>(out, copybuf);
}